// SSIM_80599356277321
// MI455X (gfx1250) — compile-verified
//
#include <hip/hip_runtime.h>
#include <math.h>

// CDNA5 / gfx1250 fused SSIM:
//  - separable 11-tap Gaussian blur as banded matmuls on V_WMMA_F32_16X16X4_F32
//  - 5 blurred channels (mu1, mu2, x1^2, x2^2, x1*x2) kept in WMMA accumulators
//  - SSIM map + mean reduction fused, single atomicAdd per workgroup
//  - branch-free border handling: clamped addresses + post-load select

typedef __attribute__((ext_vector_type(2))) float v2f;
typedef __attribute__((ext_vector_type(8))) float v8f;

#define IMG_H 512
#define IMG_W 512
#define TS    16
#define WAVES 4      // waves (== 16x16 tiles) per workgroup
#define NCH   5
#define C1v   1.0e-4f
#define C2v   9.0e-4f
#define INV_N (1.0f / 12582912.0f)   // 16*3*512*512

__device__ __forceinline__ v8f wmma_f32(v2f a, v2f b, v8f c) {
  // (neg_a, A, neg_b, B, c_mod, C, reuse_a, reuse_b) -> v_wmma_f32_16x16x4_f32
  return __builtin_amdgcn_wmma_f32_16x16x4_f32(false, a, false, b, (short)0, c,
                                               false, false);
}

__global__ __launch_bounds__(64) void ssim_zero(float* out) {
  if (threadIdx.x == 0) out[0] = 0.0f;
}

__global__ __launch_bounds__(128) void ssim_fused(const float* __restrict__ img1,
                                                  const float* __restrict__ img2,
                                                  const float* __restrict__ kern,
                                                  float* __restrict__ out) {
  __shared__ float wsh[16];                  // 1-D gaussian taps (11..15 zero)
  __shared__ float mid[WAVES][NCH][32][17];  // H-blurred tiles (stride 17: no bank conflict)
  __shared__ float red[128];                 // block reduction

  const int tid = threadIdx.x;
  // Recover 1-D taps from the 2-D kernel: K2d[i][j] = w[i]*w[j]  =>
  // w[j] = K2d[5][j] / sqrt(K2d[5][5]).  (channel 0; all channels identical)
  if (tid < 16) wsh[tid] = (tid < 11) ? (kern[55 + tid] / sqrtf(kern[60])) : 0.0f;
  __syncthreads();

  const int wave = tid >> 5;
  const int lane = tid & 31;
  const int lm = lane & 15;   // M (A) / N (B) index of this lane
  const int kh = lane >> 4;   // which K pair this half-wave holds

  const int plane = blockIdx.z;                          // 48 planes (16 batch * 3 ch)
  const int r0 = blockIdx.y * TS;                        // tile row origin
  const int c0 = (blockIdx.x * WAVES + wave) * TS;       // tile col origin
  const size_t pbase = (size_t)plane * (IMG_H * IMG_W);
  const float* p1 = img1 + pbase;
  const float* p2 = img2 + pbase;

  // Banded gaussian weight fragments, shared by H-pass B and V-pass A:
  // value at lane (lm, kh), chunk q is w[kb - lm] with kb = q*4 + kh*2
  // (in-band iff 0 <= kb-lm <= 10).  Unconditional LDS read + select: no branches.
  v2f wf[7];
#pragma unroll
  for (int q = 0; q < 7; ++q) {
    const int d0 = q * 4 + kh * 2 - lm;
    const int d1 = d0 + 1;
    const float w0 = wsh[d0 & 15];
    const float w1 = wsh[d1 & 15];
    wf[q].x = ((unsigned)d0 < 11u) ? w0 : 0.0f;
    wf[q].y = ((unsigned)d1 < 11u) ? w1 : 0.0f;
  }

  // ---------------- Horizontal pass ----------------
  // Mid(26x16) = In(26x28) * W(28x16),  W[k][n] = w[k-n] for 0<=k-n<=10.
  // M split into two 16-row WMMA tiles (rows 16..25 valid, 26..31 scratch).
#pragma unroll
  for (int t = 0; t < 2; ++t) {
    v8f acc[NCH];
#pragma unroll
    for (int ch = 0; ch < NCH; ++ch) acc[ch] = (v8f)0.0f;

    const int gr = r0 - 5 + t * 16 + lm;                 // global input row
    const bool rok = ((unsigned)gr < (unsigned)IMG_H);
    const int grc = rok ? gr : 0;                        // clamped (always in-bounds)
    const float* row1 = p1 + (size_t)grc * IMG_W;
    const float* row2 = p2 + (size_t)grc * IMG_W;

#pragma unroll
    for (int q = 0; q < 7; ++q) {
      const int kb = q * 4 + kh * 2;                     // this lane's first K
      const int gc0 = c0 - 5 + kb;                       // global input cols
      const int gc1 = gc0 + 1;
      const int gc0c = min(max(gc0, 0), IMG_W - 1);
      const int gc1c = min(max(gc1, 0), IMG_W - 1);
      // Unconditional loads from clamped addresses, then zero-select (v_cndmask):
      const float m0 = (rok && (unsigned)gc0 < (unsigned)IMG_W) ? 1.0f : 0.0f;
      const float m1 = (rok && (unsigned)gc1 < (unsigned)IMG_W) ? 1.0f : 0.0f;
      const float a0 = row1[gc0c] * m0;
      const float a1 = row1[gc1c] * m1;
      const float b0 = row2[gc0c] * m0;
      const float b1 = row2[gc1c] * m1;

      const v2f B   = wf[q];
      const v2f A1  = {a0, a1};
      const v2f A2  = {b0, b1};
      const v2f A11 = {a0 * a0, a1 * a1};
      const v2f A22 = {b0 * b0, b1 * b1};
      const v2f A12 = {a0 * b0, a1 * b1};

      acc[0] = wmma_f32(A1,  B, acc[0]);
      acc[1] = wmma_f32(A2,  B, acc[1]);
      acc[2] = wmma_f32(A11, B, acc[2]);
      acc[3] = wmma_f32(A22, B, acc[3]);
      acc[4] = wmma_f32(A12, B, acc[4]);
    }

    // D layout: VGPR v -> row v (lanes 0-15) / row v+8 (lanes 16-31), col = lane&15
#pragma unroll
    for (int v = 0; v < 8; ++v) {
      const int row = t * 16 + (kh ? (v + 8) : v);
#pragma unroll
      for (int ch = 0; ch < NCH; ++ch) mid[wave][ch][row][lm] = acc[ch][v];
    }
  }

  __syncthreads();  // cross-lane LDS visibility before vertical pass

  // ---------------- Vertical pass ----------------
  // Out(16x16) = Wv(16x28) * Mid(28x16),  Wv[m][k] = w[k-m] for 0<=k-m<=10.
  // Per-lane A values identical to wf[] (rows >=26 of Mid get zero weight).
  v8f s[NCH];
#pragma unroll
  for (int ch = 0; ch < NCH; ++ch) s[ch] = (v8f)0.0f;

#pragma unroll
  for (int q = 0; q < 7; ++q) {
    const int kb = q * 4 + kh * 2;
    const v2f A = wf[q];
#pragma unroll
    for (int ch = 0; ch < NCH; ++ch) {
      const v2f B = {mid[wave][ch][kb][lm], mid[wave][ch][kb + 1][lm]};
      s[ch] = wmma_f32(A, B, s[ch]);
    }
  }

  // ---------------- SSIM map + reduction ----------------
  float part = 0.0f;
#pragma unroll
  for (int v = 0; v < 8; ++v) {
    const float mu1 = s[0][v], mu2 = s[1][v];
    const float x11 = s[2][v], x22 = s[3][v], x12 = s[4][v];
    const float mu1s = mu1 * mu1, mu2s = mu2 * mu2, m12 = mu1 * mu2;
    const float sg1 = x11 - mu1s, sg2 = x22 - mu2s, sg12 = x12 - m12;
    const float num = (2.0f * m12 + C1v) * (2.0f * sg12 + C2v);
    const float den = (mu1s + mu2s + C1v) * (sg1 + sg2 + C2v);
    part += num / den;
  }

  red[tid] = part;
  __syncthreads();
#pragma unroll
  for (int off = 64; off > 0; off >>= 1) {
    if (tid < off) red[tid] += red[tid + off];
    __syncthreads();
  }
  if (tid == 0) atomicAdd(out, red[0] * INV_N);
}

extern "C" void kernel_launch(void* const* d_in, const int* in_sizes, int n_in,
                              void* d_out, int out_size, void* d_ws, size_t ws_size,
                              hipStream_t stream) {
  const float* img1 = (const float*)d_in[0];
  const float* img2 = (const float*)d_in[1];
  const float* kern = (const float*)d_in[2];
  float* out = (float*)d_out;

  ssim_zero<<<1, 64, 0, stream>>>(out);
  dim3 grid(IMG_W / (WAVES * TS), IMG_H / TS, 48);  // (8, 32, 48)
  dim3 block(WAVES * 32);                           // 4 waves
  ssim_fused<<<grid, block, 0, stream>>>(img1, img2, kern, out);
}